// SConv2d_13065290515016
// MI455X (gfx1250) — compile-verified
//
#include <hip/hip_runtime.h>

// SConv2d with MAJ-3 gate tree.
//   maj3(a,b,c) = (a + b + c - a*b*c) / 2   (closed form of the reference's
//   combinatorial MAJ for n=3). Tree: 27 products -> 9 -> 3 -> 1.
//
// Memory-bound problem (8 MB output vs 0.22 GFLOP) -> optimize data movement:
//  * async global->LDS staging of weights + input halo (CDNA5 ASYNCcnt path)
//  * LDS broadcast reads (wave-uniform channel group under wave32)
//  * coalesced nontemporal output stores

#define NB     8
#define CIN    3
#define COUT   64
#define HW     64
#define KDIM   27   // CIN*3*3

typedef __attribute__((address_space(1))) int* gptr_t;
typedef __attribute__((address_space(3))) int* lptr_t;

__device__ __forceinline__ float maj3(float a, float b, float c) {
    return 0.5f * (a + b + c - a * b * c);
}

__device__ __forceinline__ void async_ld_b32(const float* g, float* l) {
#if __has_builtin(__builtin_amdgcn_global_load_async_to_lds_b32)
    __builtin_amdgcn_global_load_async_to_lds_b32(
        (gptr_t)(g), (lptr_t)(l), /*offset=*/0, /*cpol=*/0);
#else
    *l = *g;   // fallback; histogram will reveal if builtin was absent
#endif
}

__device__ __forceinline__ void wait_async0() {
#if __has_builtin(__builtin_amdgcn_s_wait_asynccnt)
    __builtin_amdgcn_s_wait_asynccnt(0);
#else
    asm volatile("s_wait_asynccnt 0" ::: "memory");
#endif
}

__global__ __launch_bounds__(256) void sconv2d_maj_kernel(
    const float* __restrict__ x,      // [8][3][64][64]
    const float* __restrict__ w,      // [64][3][3][3] == [64][27]
    float* __restrict__ out)          // [8][64][64][64]
{
    __shared__ __align__(16) float wsh[COUT][28];     // padded row stride
    __shared__ __align__(16) float tile[CIN][3][66];  // 3 rows x (64+2 halo)

    const int tid = threadIdx.x;
    const int n   = blockIdx.x >> 6;   // blockIdx.x = n*64 + h
    const int h   = blockIdx.x & 63;

    // ---- zero the input tile (gives us the conv zero-padding for free) ----
    {
        float* tz = &tile[0][0][0];
        for (int idx = tid; idx < CIN * 3 * 66; idx += 256) tz[idx] = 0.0f;
    }

    // ---- async-stage weights into LDS (64*27 = 1728 b32 transfers) ----
    for (int idx = tid; idx < COUT * KDIM; idx += 256) {
        const int co = idx / KDIM;
        const int i  = idx - co * KDIM;
        async_ld_b32(w + idx, &wsh[co][i]);
    }

    __syncthreads();   // zero-stores (DScnt) complete block-wide

    // ---- async-stage input halo rows h-1..h+1 for all 3 channels ----
    for (int idx = tid; idx < CIN * 3 * 64; idx += 256) {
        const int c   = idx / 192;
        const int rem = idx - c * 192;
        const int r   = rem >> 6;
        const int wv  = rem & 63;
        const int hp  = h - 1 + r;
        if ((unsigned)hp < (unsigned)HW)
            async_ld_b32(x + (((n * CIN + c) * HW + hp) * HW + wv),
                         &tile[c][r][wv + 1]);
    }

    wait_async0();     // own wave's ASYNCcnt == 0 (weights + tile landed)
    __syncthreads();   // all waves' async writes visible

    // ---- compute: each thread = one pixel, 16 output channels ----
    const int wc = tid & 63;   // pixel column
    const int cg = tid >> 6;   // channel group 0..3 (wave-uniform -> LDS bcast)

    float p[KDIM];
#pragma unroll
    for (int c = 0; c < CIN; ++c)
#pragma unroll
        for (int r = 0; r < 3; ++r)
#pragma unroll
            for (int k = 0; k < 3; ++k)
                p[c * 9 + r * 3 + k] = tile[c][r][wc + k];

    float* op = out + (size_t)n * COUT * HW * HW + (size_t)h * HW + wc;

#pragma unroll
    for (int j = 0; j < 16; ++j) {
        const int co = cg * 16 + j;
        const float* wr = &wsh[co][0];

        float m[KDIM];
#pragma unroll
        for (int i = 0; i < KDIM; ++i) m[i] = p[i] * wr[i];

        float q[9];
#pragma unroll
        for (int t = 0; t < 9; ++t)
            q[t] = maj3(m[3 * t + 0], m[3 * t + 1], m[3 * t + 2]);

        const float v = maj3(maj3(q[0], q[1], q[2]),
                             maj3(q[3], q[4], q[5]),
                             maj3(q[6], q[7], q[8]));

        // write-once 8MB output: bypass near caches (TH=NT)
        __builtin_nontemporal_store(v, op + (size_t)co * (HW * HW));
    }
}

extern "C" void kernel_launch(void* const* d_in, const int* in_sizes, int n_in,
                              void* d_out, int out_size, void* d_ws, size_t ws_size,
                              hipStream_t stream) {
    const float* x = (const float*)d_in[0];   // 8*3*64*64 floats
    const float* w = (const float*)d_in[1];   // 64*3*3*3 floats
    float* out     = (float*)d_out;           // 8*64*64*64 floats

    sconv2d_maj_kernel<<<dim3(NB * HW), dim3(256), 0, stream>>>(x, w, out);
}